// GraphSAGE_34660386079338
// MI455X (gfx1250) — compile-verified
//
#include <hip/hip_runtime.h>
#include <math.h>

// Problem constants (fixed by the reference).
#define NN   8192   // rows / graph nodes
#define DIN  256    // in features
#define DOUT 256    // out features

typedef __attribute__((ext_vector_type(2))) float v2f;
typedef __attribute__((ext_vector_type(4))) float v4f;
typedef __attribute__((ext_vector_type(8))) float v8f;

// Workspace layout (floats):
//   [0,      8192)  deg        (init 1.0, += column sums of adj)
//   [8192,  16384)  rownorm2   (sum of squares per output row)
//   [16384, 16640)  colsum     (per-column sum of normalized h)
//   [16640, 16896)  colsumsq   (per-column sum of squares)
//   [16896, 17152)  bn scale   (written each call by k_bnprep)
//   [17152, 17408)  bn shift   (written each call by k_bnprep)
#define WS_DEG      0
#define WS_RN2      8192
#define WS_CSUM     16384
#define WS_CSQ      16640
#define WS_SCALE    16896
#define WS_SHIFT    17152
#define WS_ZERO_END 16896   // init only deg + stats; scale/shift fully overwritten

// ---------------------------------------------------------------- init
__global__ void k_init(float* __restrict__ ws) {
    int i = blockIdx.x * 256 + threadIdx.x;
    if (i < WS_ZERO_END) ws[i] = (i < WS_RN2) ? 1.0f : 0.0f;  // deg starts at +1.0
}

// ------------------------------------------------- column sums of adj
// The 256 MB bandwidth-dominant pass (~11 us at 23.3 TB/s).
// Each thread owns 4 consecutive columns -> global_load_b128, coalesced,
// non-temporal (adj is streamed once and exceeds the 192 MB L2).
// grid = (8192/1024 col-groups, 8192/128 row-blocks), block = 256.
__global__ void k_colsum(const float* __restrict__ adj, float* __restrict__ deg) {
    int c4 = blockIdx.x * 256 + threadIdx.x;       // float4 column group, 0..2047
    int r0 = blockIdx.y * 128;
    const v4f* __restrict__ p = (const v4f*)adj + (size_t)r0 * (NN / 4) + c4;
    v4f s = (v4f){};
#pragma unroll 8
    for (int r = 0; r < 128; ++r) {
        v4f t = __builtin_nontemporal_load(p);     // TH=NT streaming load
        s += t;
        p += NN / 4;
    }
    float* d = deg + c4 * 4;
    atomicAdd(d + 0, s.x);
    atomicAdd(d + 1, s.y);
    atomicAdd(d + 2, s.z);
    atomicAdd(d + 3, s.w);
}

// ------------------------------------------------------- WMMA GEMM
// out[m][n] = relu( sum_k (x[m][k]/deg[m]) * W[n][k] + b[n] )
// One wave computes a 16x64 output strip: 4 accumulator tiles of 16x16.
// V_WMMA_F32_16X16X4_F32:  A 16x4 f32 (2 VGPR), B 4x16 f32 (2 VGPR), C/D 8 VGPR.
// Per ISA layout: lanes 0-15 carry K=k0+{0,1}, lanes 16-31 carry K=k0+{2,3};
// each lane owns a single A-row (m = mstrip*16 + lane%16) for the whole K loop,
// so 1/deg is computed once per lane and folded into the A fragment.
// Also accumulates per-row sum-of-squares (for the L2 row norm) via a
// 16-lane xor-shuffle tree + one atomic per half-wave per row.
__global__ void __launch_bounds__(256)
k_gemm(const float* __restrict__ x, const float* __restrict__ W,
       const float* __restrict__ bias, const float* __restrict__ deg,
       float* __restrict__ out, float* __restrict__ rownorm2) {
    int wave   = (blockIdx.x * blockDim.x + threadIdx.x) >> 5;  // 0..2047
    int lane   = threadIdx.x & 31;
    int mstrip = wave >> 2;          // 0..511  (16 rows each)
    int nquad  = wave & 3;           // 0..3    (64 cols each)
    int half   = lane >> 4;          // K sub-block select
    int lid    = lane & 15;

    int mA = mstrip * 16 + lid;                 // this lane's A row
    float rdeg = 1.0f / deg[mA];
    const float* xrow = x + (size_t)mA * DIN;
    int nbase = nquad * 64;

    v8f acc[4];
    acc[0] = (v8f){}; acc[1] = (v8f){}; acc[2] = (v8f){}; acc[3] = (v8f){};

    for (int k0 = 0; k0 < DIN; k0 += 4) {
        int kk = k0 + half * 2;
        v2f a;
        a.x = xrow[kk]     * rdeg;
        a.y = xrow[kk + 1] * rdeg;
#pragma unroll
        for (int j = 0; j < 4; ++j) {
            int n = nbase + j * 16 + lid;       // B col; B[k][n] = W[n][k]
            const float* wrow = W + (size_t)n * DIN + kk;
            v2f bf;
            bf.x = wrow[0];
            bf.y = wrow[1];
            acc[j] = __builtin_amdgcn_wmma_f32_16x16x4_f32(
                false, a, false, bf, (short)0, acc[j], false, false);
        }
    }

    // Epilogue: bias + ReLU, store, accumulate row sum-of-squares.
#pragma unroll
    for (int j = 0; j < 4; ++j) {
        int n = nbase + j * 16 + lid;
        float bn = bias[n];
#pragma unroll
        for (int r = 0; r < 8; ++r) {
            int m = mstrip * 16 + half * 8 + r;
            float v = acc[j][r] + bn;
            v = v > 0.0f ? v : 0.0f;
            out[(size_t)m * DOUT + n] = v;
            float s = v * v;                    // reduce over the 16-lane group
            s += __shfl_xor(s, 1, 32);
            s += __shfl_xor(s, 2, 32);
            s += __shfl_xor(s, 4, 32);
            s += __shfl_xor(s, 8, 32);
            if (lid == 0) atomicAdd(rownorm2 + m, s);
        }
    }
}

// ------------------------- row L2 normalize (in place) + column stats
// block = 256 threads (one per column), each block handles 32 rows.
__global__ void k_rownorm_stats(float* __restrict__ out,
                                const float* __restrict__ rownorm2,
                                float* __restrict__ colsum,
                                float* __restrict__ colsumsq) {
    int c  = threadIdx.x;
    int r0 = blockIdx.x * 32;
    float s = 0.0f, s2 = 0.0f;
#pragma unroll 4
    for (int i = 0; i < 32; ++i) {
        int r = r0 + i;
        float inv = 1.0f / (sqrtf(rownorm2[r]) + 1e-7f);
        size_t idx = (size_t)r * DOUT + c;
        float v = out[idx] * inv;
        out[idx] = v;
        s  += v;
        s2 += v * v;
    }
    atomicAdd(colsum + c, s);
    atomicAdd(colsumsq + c, s2);
}

// ------------------------------- per-column BN scale/shift (1 block)
__global__ void k_bnprep(const float* __restrict__ colsum,
                         const float* __restrict__ colsumsq,
                         const float* __restrict__ gamma,
                         const float* __restrict__ beta,
                         float* __restrict__ scale,
                         float* __restrict__ shift) {
    int c = threadIdx.x;
    float mean = colsum[c] * (1.0f / NN);
    float var  = colsumsq[c] * (1.0f / NN) - mean * mean;   // biased, matches ref
    float sc   = gamma[c] * rsqrtf(var + 1e-5f);
    scale[c] = sc;
    shift[c] = beta[c] - mean * sc;
}

// ---------------------- BatchNorm finalize (in place, b128 streaming)
__global__ void k_bn(float* __restrict__ out,
                     const float* __restrict__ scale,
                     const float* __restrict__ shift) {
    int i4 = blockIdx.x * 256 + threadIdx.x;     // float4 index, 0..524287
    int c4 = i4 & 63;                            // float4 column group (256/4)
    v4f sc = ((const v4f*)scale)[c4];
    v4f sh = ((const v4f*)shift)[c4];
    v4f v  = ((const v4f*)out)[i4];
    v = v * sc + sh;
    ((v4f*)out)[i4] = v;
}

// ---------------------------------------------------------------- launch
extern "C" void kernel_launch(void* const* d_in, const int* in_sizes, int n_in,
                              void* d_out, int out_size, void* d_ws, size_t ws_size,
                              hipStream_t stream) {
    const float* x     = (const float*)d_in[0];   // [8192, 256]
    const float* adj   = (const float*)d_in[1];   // [8192, 8192]
    const float* W     = (const float*)d_in[2];   // [256, 256]
    const float* bias  = (const float*)d_in[3];   // [256]
    const float* gamma = (const float*)d_in[4];   // [256]
    const float* beta  = (const float*)d_in[5];   // [256]
    float* out = (float*)d_out;                   // [8192, 256]
    float* ws  = (float*)d_ws;

    float* deg      = ws + WS_DEG;
    float* rownorm2 = ws + WS_RN2;
    float* colsum   = ws + WS_CSUM;
    float* colsumsq = ws + WS_CSQ;
    float* scale    = ws + WS_SCALE;
    float* shift    = ws + WS_SHIFT;

    // 1) init scratch (deg = 1.0, stats = 0)
    k_init<<<(WS_ZERO_END + 255) / 256, 256, 0, stream>>>(ws);

    // 2) deg += column sums of adj  (256 MB NT-streamed, b128 coalesced)
    k_colsum<<<dim3(NN / 1024, NN / 128), 256, 0, stream>>>(adj, deg);

    // 3) WMMA GEMM + bias + ReLU + row-sumsq.  2048 waves = 256 blocks x 8 waves.
    k_gemm<<<256, 256, 0, stream>>>(x, W, bias, deg, out, rownorm2);

    // 4) row L2 normalize in place + column mean/var partials
    k_rownorm_stats<<<NN / 32, 256, 0, stream>>>(out, rownorm2, colsum, colsumsq);

    // 5) BN scale/shift then finalize in place (b128 streaming)
    k_bnprep<<<1, DOUT, 0, stream>>>(colsum, colsumsq, gamma, beta, scale, shift);
    k_bn<<<(NN * DOUT / 4) / 256, 256, 0, stream>>>(out, scale, shift);
}